// RelationClassifier_CLS_ent1_ent2_avg_pooled_20272245637485
// MI455X (gfx1250) — compile-verified
//
#include <hip/hip_runtime.h>
#include <hip/hip_bf16.h>

#define B_N 64
#define S_N 512
#define H_N 1024
#define K3H 3072
#define E1S_TOK 30522
#define E1E_TOK 30523
#define E2S_TOK 30524
#define E2E_TOK 30525

typedef __attribute__((ext_vector_type(2))) float v2f;
typedef __attribute__((ext_vector_type(8))) float v8f;

// ---------------------------------------------------------------------------
// Kernel 1: per-batch marker scan + span mean pooling.
// One block per batch, 256 threads; each thread owns one float4 column slot,
// so one loop trip streams exactly one 4KB row fully coalesced.
// Writes combined[b][0:3072] = [cls | ent1_mean | ent2_mean].
// ---------------------------------------------------------------------------
__global__ __launch_bounds__(256) void span_pool_kernel(
    const int* __restrict__ ids, const float* __restrict__ seq,
    float* __restrict__ combined)
{
    const int b   = blockIdx.x;
    const int tid = threadIdx.x;

    __shared__ int pos[4];
    if (tid < 4) pos[tid] = S_N;      // "not found" sentinel
    __syncthreads();

    const int* row_ids = ids + (size_t)b * S_N;
    for (int t = tid; t < S_N; t += 256) {
        int tok = row_ids[t];
        if      (tok == E1S_TOK) atomicMin(&pos[0], t);
        else if (tok == E1E_TOK) atomicMin(&pos[1], t);
        else if (tok == E2S_TOK) atomicMin(&pos[2], t);
        else if (tok == E2E_TOK) atomicMin(&pos[3], t);
    }
    __syncthreads();

    const int s1 = pos[0], e1 = pos[1], s2 = pos[2], e2 = pos[3];
    const bool f1 = (s1 < S_N) && (e1 < S_N);
    const bool f2 = (s2 < S_N) && (e2 < S_N);
    const int st1 = s1 + 1, en1 = f1 ? e1 : 0;
    const int st2 = s2 + 1, en2 = f2 ? e2 : 0;
    const int cnt1 = (f1 && en1 > st1) ? (en1 - st1) : 0;
    const int cnt2 = (f2 && en2 > st2) ? (en2 - st2) : 0;

    const float4* srow = (const float4*)(seq + (size_t)b * S_N * H_N);
    const int HV = H_N / 4;   // 256 float4 per row == blockDim

    // CLS row (row 0)
    float4 cls = srow[tid];

    // span 1 sum
    float4 a1; a1.x = a1.y = a1.z = a1.w = 0.0f;
    for (int r = st1; r < en1; ++r) {
        float4 v = srow[(size_t)r * HV + tid];
        a1.x += v.x; a1.y += v.y; a1.z += v.z; a1.w += v.w;
    }
    // span 2 sum
    float4 a2; a2.x = a2.y = a2.z = a2.w = 0.0f;
    for (int r = st2; r < en2; ++r) {
        float4 v = srow[(size_t)r * HV + tid];
        a2.x += v.x; a2.y += v.y; a2.z += v.z; a2.w += v.w;
    }

    const float inv1 = (cnt1 > 0) ? (1.0f / (float)cnt1) : 0.0f;
    const float inv2 = (cnt2 > 0) ? (1.0f / (float)cnt2) : 0.0f;
    a1.x *= inv1; a1.y *= inv1; a1.z *= inv1; a1.w *= inv1;
    a2.x *= inv2; a2.y *= inv2; a2.z *= inv2; a2.w *= inv2;

    float4* out = (float4*)(combined + (size_t)b * K3H);
    out[tid]            = cls;   // cols    0..1023
    out[HV + tid]       = a1;    // cols 1024..2047
    out[2 * HV + tid]   = a2;    // cols 2048..3071
}

// ---------------------------------------------------------------------------
// Kernel 2: logits = combined(64x3072) @ W(3072) + b via V_WMMA_F32_16X16X4_F32
// (full fp32, matches reference precision). One wave per 16-batch M-tile.
// A layout (ISA 7.12.2, 32-bit 16x4): lane L holds row M=(L&15);
//   VGPR0 = K=(L<16?0:2), VGPR1 = K=(L<16?1:3)  -> per-lane float2 at
//   offset kc + (L>>4)*2. B (4x16) mirrored; broadcast W chunk across all N
//   columns so every D column equals the logit.
// ---------------------------------------------------------------------------
__global__ __launch_bounds__(32) void gemv_wmma_kernel(
    const float* __restrict__ combined, const float* __restrict__ W,
    const float* __restrict__ bias, float* __restrict__ out)
{
    const int lane = threadIdx.x;
    const int mt   = blockIdx.x;
    const int row  = mt * 16 + (lane & 15);
    const int kh   = (lane >> 4) * 2;            // 0 or 2

    const float* arow = combined + (size_t)row * K3H + kh;
    const float* wrow = W + kh;

    v8f c0 = {};
    v8f c1 = {};
    #pragma unroll 4
    for (int kc = 0; kc < K3H; kc += 8) {
        v2f a0 = *(const v2f*)(arow + kc);
        v2f b0 = *(const v2f*)(wrow + kc);
        c0 = __builtin_amdgcn_wmma_f32_16x16x4_f32(
                false, a0, false, b0, (short)0, c0, false, false);
        v2f a1 = *(const v2f*)(arow + kc + 4);
        v2f b1 = *(const v2f*)(wrow + kc + 4);
        c1 = __builtin_amdgcn_wmma_f32_16x16x4_f32(
                false, a1, false, b1, (short)0, c1, false, false);
    }

    const float bs = bias[0];
    // D layout: VGPR i, lanes 0-15 -> M=i ; lanes 16-31 -> M=8+i (all N equal).
    if ((lane & 15) == 0) {
        const int base = mt * 16 + (lane >> 4) * 8;
        out[base + 0] = c0[0] + c1[0] + bs;
        out[base + 1] = c0[1] + c1[1] + bs;
        out[base + 2] = c0[2] + c1[2] + bs;
        out[base + 3] = c0[3] + c1[3] + bs;
        out[base + 4] = c0[4] + c1[4] + bs;
        out[base + 5] = c0[5] + c1[5] + bs;
        out[base + 6] = c0[6] + c1[6] + bs;
        out[base + 7] = c0[7] + c1[7] + bs;
    }
}

// ---------------------------------------------------------------------------
// Fallback (if workspace too small): fully fused span-find + direct dot.
// ---------------------------------------------------------------------------
__global__ __launch_bounds__(256) void fused_direct_kernel(
    const int* __restrict__ ids, const float* __restrict__ seq,
    const float* __restrict__ W, const float* __restrict__ bias,
    float* __restrict__ out)
{
    const int b   = blockIdx.x;
    const int tid = threadIdx.x;

    __shared__ int pos[4];
    __shared__ float red[256];
    if (tid < 4) pos[tid] = S_N;
    __syncthreads();

    const int* row_ids = ids + (size_t)b * S_N;
    for (int t = tid; t < S_N; t += 256) {
        int tok = row_ids[t];
        if      (tok == E1S_TOK) atomicMin(&pos[0], t);
        else if (tok == E1E_TOK) atomicMin(&pos[1], t);
        else if (tok == E2S_TOK) atomicMin(&pos[2], t);
        else if (tok == E2E_TOK) atomicMin(&pos[3], t);
    }
    __syncthreads();

    const int s1 = pos[0], e1 = pos[1], s2 = pos[2], e2 = pos[3];
    const bool f1 = (s1 < S_N) && (e1 < S_N);
    const bool f2 = (s2 < S_N) && (e2 < S_N);
    const int st1 = s1 + 1, en1 = f1 ? e1 : 0;
    const int st2 = s2 + 1, en2 = f2 ? e2 : 0;
    const int cnt1 = (f1 && en1 > st1) ? (en1 - st1) : 0;
    const int cnt2 = (f2 && en2 > st2) ? (en2 - st2) : 0;

    const float4* srow = (const float4*)(seq + (size_t)b * S_N * H_N);
    const float4* w0 = (const float4*)W;
    const int HV = H_N / 4;

    float4 cls = srow[tid];
    float4 a1; a1.x = a1.y = a1.z = a1.w = 0.0f;
    for (int r = st1; r < en1; ++r) {
        float4 v = srow[(size_t)r * HV + tid];
        a1.x += v.x; a1.y += v.y; a1.z += v.z; a1.w += v.w;
    }
    float4 a2; a2.x = a2.y = a2.z = a2.w = 0.0f;
    for (int r = st2; r < en2; ++r) {
        float4 v = srow[(size_t)r * HV + tid];
        a2.x += v.x; a2.y += v.y; a2.z += v.z; a2.w += v.w;
    }
    const float inv1 = (cnt1 > 0) ? (1.0f / (float)cnt1) : 0.0f;
    const float inv2 = (cnt2 > 0) ? (1.0f / (float)cnt2) : 0.0f;

    float4 wc = w0[tid];
    float4 w1 = w0[HV + tid];
    float4 w2 = w0[2 * HV + tid];

    float partial =
        cls.x * wc.x + cls.y * wc.y + cls.z * wc.z + cls.w * wc.w +
        inv1 * (a1.x * w1.x + a1.y * w1.y + a1.z * w1.z + a1.w * w1.w) +
        inv2 * (a2.x * w2.x + a2.y * w2.y + a2.z * w2.z + a2.w * w2.w);

    red[tid] = partial;
    __syncthreads();
    for (int s = 128; s > 0; s >>= 1) {
        if (tid < s) red[tid] += red[tid + s];
        __syncthreads();
    }
    if (tid == 0) out[b] = red[0] + bias[0];
}

// ---------------------------------------------------------------------------
extern "C" void kernel_launch(void* const* d_in, const int* in_sizes, int n_in,
                              void* d_out, int out_size, void* d_ws, size_t ws_size,
                              hipStream_t stream) {
    const int*   ids  = (const int*)d_in[0];
    // d_in[1] = attention_mask (unused by reference math)
    const float* seq  = (const float*)d_in[2];
    const float* W    = (const float*)d_in[3];
    const float* bias = (const float*)d_in[4];
    float*       out  = (float*)d_out;

    const size_t needed = (size_t)B_N * K3H * sizeof(float);
    if (d_ws != nullptr && ws_size >= needed) {
        float* combined = (float*)d_ws;
        span_pool_kernel<<<B_N, 256, 0, stream>>>(ids, seq, combined);
        gemv_wmma_kernel<<<B_N / 16, 32, 0, stream>>>(combined, W, bias, out);
    } else {
        fused_direct_kernel<<<B_N, 256, 0, stream>>>(ids, seq, W, bias, out);
    }
}